// LoLAState_54073638257061
// MI455X (gfx1250) — compile-verified
//
#include <hip/hip_runtime.h>
#include <stdint.h>

typedef __attribute__((ext_vector_type(2))) float v2f;
typedef __attribute__((ext_vector_type(8))) float v8f;

// Problem dimensions (fixed by the reference).
constexpr int B = 2, H = 32, C = 256, D = 128, G = 2048, F = 128;
constexpr int BH = B * H;
constexpr int NCAND = G + C;   // 2304 top-k candidates
constexpr int MSORT = 4096;    // padded bitonic size
constexpr int CH = 64;         // g-rows per LDS chunk in hsum kernel
constexpr int NCHUNK = G / CH; // 32

// Flat output offsets (floats), in reference return order.
constexpr long long o_Kwin  = 0;
constexpr long long o_Vwin  = o_Kwin  + (long long)BH * C * D;
constexpr long long o_Ktop  = o_Vwin  + (long long)BH * C * D;
constexpr long long o_Vtop  = o_Ktop  + (long long)BH * G * D;
constexpr long long o_FKtop = o_Vtop  + (long long)BH * G * D;
constexpr long long o_nval  = o_FKtop + (long long)BH * G * F;
constexpr long long o_nidx  = o_nval  + (long long)BH * G;
constexpr long long o_Hsum  = o_nidx  + (long long)BH * G;
constexpr long long o_Ssum  = o_Hsum  + (long long)BH * F * D;

// --------------------------- async-to-LDS support --------------------------
#if defined(__AMDGCN__) && __has_builtin(__builtin_amdgcn_global_load_async_to_lds_b128)
#define ASYNC_LDS 1
#else
#define ASYNC_LDS 0
#endif

#if defined(__AMDGCN__) && __has_builtin(__builtin_amdgcn_s_wait_asynccnt)
#define WAIT_ASYNC(n) __builtin_amdgcn_s_wait_asynccnt(n)
#else
#define WAIT_ASYNC(n) asm volatile("s_wait_asynccnt %0" ::"n"(n) : "memory")
#endif

#if ASYNC_LDS
// Builtin parameter types per the compiler diagnostic:
//   arg0: 'int __vector(4) __device__ *'  == AS(1) pointer to gcc int4
//   arg1: LDS-side pointer (AS(3)) to the same vector type
typedef int v4i_gcc __attribute__((__vector_size__(16)));
typedef __attribute__((address_space(1))) v4i_gcc as1_v4i;
typedef __attribute__((address_space(3))) v4i_gcc as3_v4i;

// Issue one 64-row x 128-col fp32 chunk (32 KB) per array as async b128 loads.
// 8 instructions per thread per array -> 16 ASYNCcnt increments per wave.
__device__ __forceinline__ void async_issue_chunk(const float* gFK, const float* gV,
                                                  float* sFK, float* sV, int tid) {
    uintptr_t gf = (uintptr_t)gFK, gv = (uintptr_t)gV;
    uint32_t  lf = (uint32_t)(uintptr_t)sFK;   // flat LDS addr low 32b == LDS offset
    uint32_t  lv = (uint32_t)(uintptr_t)sV;
#pragma unroll
    for (int v = 0; v < 8; ++v) {
        uint32_t byt = (uint32_t)(tid + v * 256) * 16u;
        __builtin_amdgcn_global_load_async_to_lds_b128(
            (as1_v4i*)(gf + byt), (as3_v4i*)(uintptr_t)(lf + byt), 0, 0);
        __builtin_amdgcn_global_load_async_to_lds_b128(
            (as1_v4i*)(gv + byt), (as3_v4i*)(uintptr_t)(lv + byt), 0, 0);
    }
}
#endif

// ---------------------------------------------------------------------------
// Kernel 1: sliding-window shift. K_win_new[:, :, :-1] = K_win[:, :, 1:],
// last slot = k_c[:, :, C-1]. Same for V. Pure float4 streaming copy.
// ---------------------------------------------------------------------------
__global__ void window_shift(const float4* __restrict__ Kwin,
                             const float4* __restrict__ Vwin,
                             const float4* __restrict__ kc,
                             const float4* __restrict__ vc,
                             float4* __restrict__ out4) {
    const int n4 = BH * C * (D / 4);
    int id = blockIdx.x * blockDim.x + threadIdx.x;
    if (id >= 2 * n4) return;
    bool isV = (id >= n4);
    int t = isV ? id - n4 : id;
    int bh  = t / (C * (D / 4));
    int rem = t - bh * (C * (D / 4));
    int i   = rem / (D / 4);
    int d4  = rem - i * (D / 4);
    const float4* win = isV ? Vwin : Kwin;
    const float4* cc  = isV ? vc : kc;
    float4 v;
    if (i < C - 1) v = win[((long long)bh * C + i + 1) * (D / 4) + d4];
    else           v = cc[((long long)bh * C + (C - 1)) * (D / 4) + d4];
    out4[id] = v;   // K region at o_Kwin, V region immediately after (o_Vwin)
}

// ---------------------------------------------------------------------------
// Kernel 2: per-(b,h) top-G of [heap_val | score_c] via in-LDS bitonic sort of
// 64-bit keys: (orderable_float << 32) | ~idx  -> sort descending gives
// descending values with ascending-index tie-break (top_k stability).
// ---------------------------------------------------------------------------
__global__ __launch_bounds__(256) void topk_sort(
    const float* __restrict__ heap_val, const float* __restrict__ score_c,
    const int* __restrict__ heap_idx, const int* __restrict__ tokens_seen_p,
    float* __restrict__ out, int* __restrict__ sel_ws) {
    __shared__ unsigned long long key[MSORT];   // 32 KB
    const int bh  = blockIdx.x;
    const int tid = threadIdx.x;

    for (int i = tid; i < MSORT; i += 256) {
        unsigned long long k = 0ull;   // padding sorts last (all real keys > 0)
        if (i < NCAND) {
            float v = (i < G) ? heap_val[(long long)bh * G + i]
                              : score_c[(long long)bh * C + (i - G)];
            unsigned u = __float_as_uint(v);
            u = (u & 0x80000000u) ? ~u : (u | 0x80000000u);  // total order, asc
            k = ((unsigned long long)u << 32) | (unsigned)(~(unsigned)i);
        }
        key[i] = k;
    }
    __syncthreads();

    for (int kk = 2; kk <= MSORT; kk <<= 1) {
        for (int j = kk >> 1; j > 0; j >>= 1) {
            for (int t = tid; t < MSORT / 2; t += 256) {
                int i = ((t & ~(j - 1)) << 1) | (t & (j - 1));
                int p = i | j;
                unsigned long long a = key[i], b = key[p];
                bool descBlock = ((i & kk) == 0);
                if ((a < b) == descBlock) { key[i] = b; key[p] = a; }
            }
            __syncthreads();
        }
    }

    const int tokens_seen = tokens_seen_p[0];
    float* nval = out + o_nval + (long long)bh * G;
    int*   nidx = (int*)(out + o_nidx) + (long long)bh * G;
    for (int g = tid; g < G; g += 256) {
        unsigned long long k = key[g];
        unsigned tv = (unsigned)(k >> 32);
        unsigned u  = (tv & 0x80000000u) ? (tv & 0x7fffffffu) : ~tv;
        int sel = (int)(~(unsigned)(k & 0xffffffffull));
        nval[g] = __uint_as_float(u);
        nidx[g] = (sel < G) ? heap_idx[(long long)bh * G + sel]
                            : (tokens_seen + (sel - G));
        sel_ws[(long long)bh * G + g] = sel;
    }
}

// ---------------------------------------------------------------------------
// Kernel 3: gather surviving rows of K/V/FK from [old top | new chunk].
// One block per (b,h,g) row; 128 threads copy 3 contiguous 512B rows.
// ---------------------------------------------------------------------------
__global__ __launch_bounds__(128) void gather_rows(
    const float* __restrict__ K_top, const float* __restrict__ V_top,
    const float* __restrict__ FK_top, const float* __restrict__ kc,
    const float* __restrict__ vc, const float* __restrict__ fkc,
    const int* __restrict__ sel_ws, float* __restrict__ out) {
    const int bhg = blockIdx.x;        // 0 .. BH*G-1
    const int bh  = bhg / G;
    const int t   = threadIdx.x;       // 0..127
    const int s   = sel_ws[bhg];
    const float *srcK, *srcV, *srcFK;
    if (s < G) {
        long long r = (long long)bh * G + s;
        srcK = K_top + r * D; srcV = V_top + r * D; srcFK = FK_top + r * F;
    } else {
        long long r = (long long)bh * C + (s - G);
        srcK = kc + r * D; srcV = vc + r * D; srcFK = fkc + r * F;
    }
    long long o = (long long)bhg;
    out[o_Ktop  + o * D + t] = srcK[t];
    out[o_Vtop  + o * D + t] = srcV[t];
    out[o_FKtop + o * F + t] = srcFK[t];
}

// ---------------------------------------------------------------------------
// Kernel 4: H_sum[b,h,f,d] = sum_g FK[g,f] * V[g,d]  via V_WMMA_F32_16X16X4_F32
// (fp32 A/B -> exact precision). One block (8 wave32) per (b,h). G staged
// through double-buffered LDS chunks fed by GLOBAL_LOAD_ASYNC_TO_LDS_B128
// (ASYNCcnt-tracked) when available. Wave w owns f-block w and all 8 d-blocks,
// reusing each A fragment across the 8 WMMAs. S_sum folded in.
// ---------------------------------------------------------------------------
struct HsumCtx {
    int tid, hl, l, f0;
    v8f acc[8];
    float ss;
};

__device__ __forceinline__ void hsum_compute(HsumCtx& cx, const float* sFK,
                                             const float* sV) {
    for (int kk = 0; kk < CH; kk += 4) {
        int r0 = kk + 2 * cx.hl;      // this half-wave's K rows (r0, r0+1)
        v2f a;
        a.x = sFK[(r0 + 0) * 128 + cx.f0 + cx.l];
        a.y = sFK[(r0 + 1) * 128 + cx.f0 + cx.l];
#pragma unroll
        for (int db = 0; db < 8; ++db) {
            v2f bf;
            bf.x = sV[(r0 + 0) * 128 + db * 16 + cx.l];
            bf.y = sV[(r0 + 1) * 128 + db * 16 + cx.l];
            cx.acc[db] = __builtin_amdgcn_wmma_f32_16x16x4_f32(
                false, a, false, bf, (short)0, cx.acc[db], false, false);
        }
    }
    if (cx.tid < 128) {               // S_sum partial from same LDS tile
        for (int g = 0; g < CH; ++g) cx.ss += sFK[g * 128 + cx.tid];
    }
}

__global__ __launch_bounds__(256) void hsum_wmma(float* __restrict__ out) {
#if ASYNC_LDS
    __shared__ float sFK[2][CH * 128];   // 2 x 32 KB
    __shared__ float sV [2][CH * 128];   // 2 x 32 KB
#else
    __shared__ float sFK[1][CH * 128];
    __shared__ float sV [1][CH * 128];
#endif
    const int bh   = blockIdx.x;
    const int tid  = threadIdx.x;
    const int lane = tid & 31;

    HsumCtx cx;
    cx.tid = tid;
    cx.hl  = lane >> 4;
    cx.l   = lane & 15;
    cx.f0  = (tid >> 5) * 16;         // wave id -> f-block
#pragma unroll
    for (int i = 0; i < 8; ++i) cx.acc[i] = v8f{};
    cx.ss = 0.0f;

    const float* FKt = out + o_FKtop + (long long)bh * G * F;
    const float* Vt  = out + o_Vtop  + (long long)bh * G * D;

#if ASYNC_LDS
    async_issue_chunk(FKt, Vt, sFK[0], sV[0], tid);          // prefetch chunk 0
    for (int c = 0; c < NCHUNK; ++c) {
        const int cur = c & 1;
        if (c + 1 < NCHUNK) {
            async_issue_chunk(FKt + (long long)(c + 1) * CH * F,
                              Vt  + (long long)(c + 1) * CH * D,
                              sFK[cur ^ 1], sV[cur ^ 1], tid);
            WAIT_ASYNC(16);           // 16 oldest (= chunk c) complete, in order
        } else {
            WAIT_ASYNC(0);
        }
        __syncthreads();
        hsum_compute(cx, sFK[cur], sV[cur]);
        __syncthreads();              // all reads done before buf is re-filled
    }
#else
    for (int c = 0; c < NCHUNK; ++c) {
        const float4* gFK = (const float4*)(FKt + (long long)c * CH * F);
        const float4* gV  = (const float4*)(Vt  + (long long)c * CH * D);
        float4* lFK = (float4*)sFK[0];
        float4* lV  = (float4*)sV[0];
        for (int v = tid; v < CH * 128 / 4; v += 256) {  // uniform trip count
            lFK[v] = gFK[v];
            lV[v]  = gV[v];
        }
        __syncthreads();
        hsum_compute(cx, sFK[0], sV[0]);
        __syncthreads();
    }
#endif

    // Store: C/D layout -> VGPR r holds rows (r | r+8), lanes 0-15 | 16-31.
    float* Hs = out + o_Hsum + (long long)bh * F * D;
#pragma unroll
    for (int db = 0; db < 8; ++db)
#pragma unroll
        for (int r = 0; r < 8; ++r)
            Hs[(cx.f0 + r + 8 * cx.hl) * D + db * 16 + cx.l] = cx.acc[db][r];

    if (tid < 128) out[o_Ssum + (long long)bh * F + tid] = cx.ss;
}

// ---------------------------------------------------------------------------
extern "C" void kernel_launch(void* const* d_in, const int* in_sizes, int n_in,
                              void* d_out, int out_size, void* d_ws, size_t ws_size,
                              hipStream_t stream) {
    const float* k_c      = (const float*)d_in[0];
    const float* v_c      = (const float*)d_in[1];
    const float* fk_c     = (const float*)d_in[2];
    const float* score_c  = (const float*)d_in[3];
    const float* K_win    = (const float*)d_in[4];
    const float* V_win    = (const float*)d_in[5];
    const float* K_top    = (const float*)d_in[6];
    const float* V_top    = (const float*)d_in[7];
    const float* FK_top   = (const float*)d_in[8];
    const float* heap_val = (const float*)d_in[9];
    const int*   heap_idx = (const int*)d_in[10];
    const int*   tokens   = (const int*)d_in[11];
    float* out   = (float*)d_out;
    int*   selws = (int*)d_ws;       // BH*G ints = 512 KB

    const int n4 = BH * C * (D / 4);
    window_shift<<<(2 * n4 + 255) / 256, 256, 0, stream>>>(
        (const float4*)K_win, (const float4*)V_win,
        (const float4*)k_c, (const float4*)v_c, (float4*)out);

    topk_sort<<<BH, 256, 0, stream>>>(heap_val, score_c, heap_idx, tokens,
                                      out, selws);

    gather_rows<<<BH * G, 128, 0, stream>>>(K_top, V_top, FK_top,
                                            k_c, v_c, fk_c, selws, out);

    hsum_wmma<<<BH, 256, 0, stream>>>(out);
}